// SpikingNeuron_61821759259043
// MI455X (gfx1250) — compile-verified
//
#include <hip/hip_runtime.h>
#include <cstdint>
#include <cstddef>

// LIF scan: B=16, S=2048, H=1024, fp32.
// One thread per (b,h) channel; sequential over s with a DEPTH-deep
// async global->LDS ring buffer per wave (CDNA5 ASYNCcnt path).
//
// Roofline: 268 MB of traffic -> ~11.5 us @ 23.3 TB/s. Parallelism capped at
// B*H = 16384 lanes = 512 wave32s, so latency hiding comes from the per-wave
// async pipeline (DEPTH outstanding 128B lines/wave) and a short dependent
// chain (fma -> fma -> v_tanh -> fma per timestep).

#define TPB   128   // threads per block (4 waves)
#define DEPTH 32    // async pipeline stages (<= 63 ASYNCcnt)

static constexpr int Bn = 16;
static constexpr int Sn = 2048;
static constexpr int Hn = 1024;

// ---- CDNA5 async-copy + wait plumbing ----

#if defined(__has_builtin)
#  if __has_builtin(__builtin_amdgcn_s_wait_asynccnt)
#    define WAIT_ASYNC(n) __builtin_amdgcn_s_wait_asynccnt(n)
#  endif
#endif
#ifndef WAIT_ASYNC
#  define WAIT_ASYNC(n) asm volatile("s_wait_asynccnt %0" :: "n"(n) : "memory")
#endif

__device__ __forceinline__ void compiler_fence() { asm volatile("" ::: "memory"); }

typedef __attribute__((address_space(1))) int as1_int;  // global
typedef __attribute__((address_space(3))) int as3_int;  // LDS

__device__ __forceinline__ void async_copy_b32(float* lds_dst, const float* gsrc) {
#if defined(__has_builtin) && __has_builtin(__builtin_amdgcn_global_load_async_to_lds_b32)
  as1_int* g = (as1_int*)(uintptr_t)(const void*)gsrc;   // global: bit-identical reinterp
  as3_int* l = (as3_int*)(int*)lds_dst;                  // LDS: addrspacecast generic->AS3
  __builtin_amdgcn_global_load_async_to_lds_b32(g, l, /*offset=*/0, /*cpol=*/0);
#else
  unsigned int lo = (unsigned int)(uintptr_t)(as3_int*)(int*)lds_dst;
  asm volatile("global_load_async_to_lds_b32 %0, %1, off"
               :: "v"(lo), "v"(gsrc) : "memory");
#endif
}

// ---- LIF step math ----
// Reference: m = m*decay + cur; o = sigmoid((m - th)*5); m = m - 0.5*o*m.
// With u = tanh(2.5*m - 2.5*th):  o = 0.5*u + 0.5,  m_new = 0.75*m - 0.25*u*m.
// Critical chain: fma(decay) -> fma(w) -> tanh -> fma(m_new); a=0.25m, 3a and o
// are off the recurrence path.

__device__ __forceinline__ float lif_step(float m, float cur, float decay,
                                          float thn /* = -2.5*th */, float& o) {
  m = fmaf(m, decay, cur);              // mem = mem*decay + cur
  const float w = fmaf(m, 2.5f, thn);   // 2.5*(mem - th)
#if defined(__has_builtin) && __has_builtin(__builtin_amdgcn_tanhf)
  const float u = __builtin_amdgcn_tanhf(w);   // single hw TRANS op
#else
  const float e = __expf(-2.0f * w);
  const float u = fmaf(-2.0f, __builtin_amdgcn_rcpf(1.0f + e), 1.0f); // -tanh(w)
#endif
  const float a = 0.25f * m;            // off-path (parallel with tanh)
  const float b = 3.0f * a;             // 0.75*m, off-path
  o = fmaf(u, 0.5f, 0.5f);              // output, off the recurrence path
  return fmaf(u, -a, b);                // m_new = 0.75m - 0.25*u*m
}

__global__ __launch_bounds__(TPB, 1) void lif_scan_kernel(
    const float* __restrict__ in,     // [B,S,H]
    const float* __restrict__ thr_p,  // [H]
    const float* __restrict__ dec_p,  // [H]
    float* __restrict__ out)          // [B*S*H] outs ++ [B*H] mem_final
{
  __shared__ float stage[DEPTH * TPB];   // 16 KB ring, per-wave disjoint slices

  const int tid = threadIdx.x;
  const int c   = blockIdx.x * TPB + tid;   // channel id 0..B*H-1
  const int b   = c >> 10;                  // / H
  const int h   = c & (Hn - 1);             // % H

  // Per-channel constants (computed once).
  const float dp = dec_p[h];
  const float tp = thr_p[h];
  const float decay = 1.0f / (1.0f + expf(-dp));                    // sigmoid
  const float th    = fmaxf(tp, 0.0f) + log1pf(expf(-fabsf(tp)));   // softplus
  const float thn   = -2.5f * th;

  const size_t base = ((size_t)b * Sn) * Hn + (size_t)h;
  const float* gin  = in  + base;
  float*       gout = out + base;
  float*       lme  = &stage[tid];          // my element in stage 0

  // Prologue: fill the async pipeline (DEPTH outstanding per wave).
#pragma unroll
  for (int i = 0; i < DEPTH; ++i)
    async_copy_b32(lme + i * TPB, gin + (size_t)i * Hn);

  const float* gpf = gin + (size_t)DEPTH * Hn;   // next address to prefetch
  float m = 0.0f;
  int s = 0;

  // Steady state: consume oldest stage, refill it with step s+DEPTH.
  for (; s < Sn - DEPTH; ++s) {
    WAIT_ASYNC(DEPTH - 1);                  // load for step s has landed
    compiler_fence();
    const int st = s & (DEPTH - 1);
    const float cur = stage[st * TPB + tid];
    async_copy_b32(lme + st * TPB, gpf);
    gpf += Hn;
    float o;
    m = lif_step(m, cur, decay, thn, o);
    __builtin_nontemporal_store(o, gout);
    gout += Hn;
  }

  // Drain: everything has been issued; wait for all, then finish the tail.
  WAIT_ASYNC(0);
  compiler_fence();
#pragma unroll
  for (int t = 0; t < DEPTH; ++t, ++s) {
    const int st = s & (DEPTH - 1);
    const float cur = stage[st * TPB + tid];
    float o;
    m = lif_step(m, cur, decay, thn, o);
    __builtin_nontemporal_store(o, gout);
    gout += Hn;
  }

  // Final membrane state.
  out[(size_t)Bn * Sn * Hn + (size_t)b * Hn + h] = m;
}

extern "C" void kernel_launch(void* const* d_in, const int* in_sizes, int n_in,
                              void* d_out, int out_size, void* d_ws, size_t ws_size,
                              hipStream_t stream) {
  const float* input = (const float*)d_in[0];   // [B,S,H]
  const float* thr   = (const float*)d_in[1];   // [H]
  const float* dec   = (const float*)d_in[2];   // [H]
  float* out = (float*)d_out;

  const int channels = Bn * Hn;                 // 16384
  dim3 grid(channels / TPB), block(TPB);
  lif_scan_kernel<<<grid, block, 0, stream>>>(input, thr, dec, out);
}